// ST2_5248450036348
// MI455X (gfx1250) — compile-verified
//
#include <hip/hip_runtime.h>
#include <cstdint>
#include <cstddef>

#define ROW 128
#define COL 128
#define NPIX (ROW * COL)
#define THREADS 1024
#define PIX_PER_THREAD (NPIX / THREADS)   // 16, one contiguous row segment

// Affine-grid bilinear sampler (spatial transformer).
// grid = (chan, batch); one workgroup per (b, c) image.
// The 64 KB image is staged into LDS with CDNA5 async global->LDS copies
// (ASYNCcnt / s_wait_asynccnt), then each thread does 4-tap bilinear
// gathers from LDS and writes its 16 contiguous pixels as 4x b128 stores.
__global__ __launch_bounds__(THREADS) void st_bilinear_kernel(
    const float* __restrict__ X,         // (b, chan, 128, 128)
    const float* __restrict__ theta_raw, // (b, 6)
    float* __restrict__ out)             // (b, chan, 128, 128)
{
    __shared__ float img[NPIX];  // 65536 bytes of the 320 KB WGP LDS

    const int c    = blockIdx.x;
    const int b    = blockIdx.y;
    const int chan = gridDim.x;
    const int bc   = b * chan + c;
    const int tid  = threadIdx.x;

    // ---------- stage the whole (b,c) image into LDS (async path) ----------
    const uint64_t sbase    = (uint64_t)(uintptr_t)(X + (size_t)bc * NPIX);
    const unsigned lds_base = (unsigned)(uintptr_t)(&img[0]);

    // 64 KB = 4096 x 16-byte chunks; 1024 threads -> 4 chunks per thread.
    #pragma unroll
    for (int k = 0; k < NPIX * 4 / (16 * THREADS); ++k) {
        const unsigned goff  = (unsigned)(tid + k * THREADS) * 16u;
        const unsigned laddr = lds_base + goff;
        // GVS mode: copies 16 bytes per lane from global sbase + goff to the LDS address in laddr, bumps ASYNCcnt
        asm volatile("global_load_async_to_lds_b128 %0, %1, %2 offset:0"
                     :
                     : "v"(laddr), "v"(goff), "s"(sbase)
                     : "memory");
    }
    asm volatile("s_wait_asynccnt 0" ::: "memory");  // my wave's copies landed
    __syncthreads();                                 // everyone's copies landed

    // ---------- per-image affine parameters (uniform; scalar loads) --------
    const float* tr = theta_raw + b * 6;
    const float s0 = 1.0f / (1.0f + expf(-tr[0]));        // theta[0][0]
    const float s1 = 1.0f / (1.0f + expf(-tr[1]));        // theta[0][1]
    const float s2 = 1.0f / (1.0f + expf(-tr[2]));        // theta[1][0]
    const float s3 = 1.0f / (1.0f + expf(-tr[3]));        // theta[1][1]
    const float t0 = (float)ROW / (1.0f + expf(-tr[4]));  // theta[0][2]
    const float t1 = (float)ROW / (1.0f + expf(-tr[5]));  // theta[1][2]

    // ---------- each thread: one 16-pixel row segment -----------------------
    const int   p     = tid >> 3;          // output row (0..127)
    const int   q0    = (tid & 7) << 4;    // first output col of the segment
    const float fp    = (float)p;
    const float basex = s1 * fp + t0;      // gx = s0*q + basex
    const float basey = s3 * fp + t1;      // gy = s2*q + basey

    float4* ovec =
        (float4*)(out + (size_t)bc * NPIX + ((size_t)p << 7) + q0);

    #pragma unroll
    for (int g = 0; g < PIX_PER_THREAD / 4; ++g) {
        float4 r;
        float* rp = &r.x;
        #pragma unroll
        for (int j = 0; j < 4; ++j) {
            const float fq = (float)(q0 + g * 4 + j);
            const float gx = s0 * fq + basex;   // sample col coordinate
            const float gy = s2 * fq + basey;   // sample row coordinate

            const int   x0 = (int)floorf(gx);
            const int   y0 = (int)floorf(gy);
            const float fx = gx - (float)x0;
            const float fy = gy - (float)y0;

            const bool x0ok = (unsigned)x0       < (unsigned)COL;
            const bool x1ok = (unsigned)(x0 + 1) < (unsigned)COL;
            const bool y0ok = (unsigned)y0       < (unsigned)ROW;
            const bool y1ok = (unsigned)(y0 + 1) < (unsigned)ROW;

            // hat weights, zeroed for out-of-range taps (branch-free)
            const float w00 = (x0ok && y0ok) ? (1.0f - fx) * (1.0f - fy) : 0.0f;
            const float w10 = (x1ok && y0ok) ? fx * (1.0f - fy)          : 0.0f;
            const float w01 = (x0ok && y1ok) ? (1.0f - fx) * fy          : 0.0f;
            const float w11 = (x1ok && y1ok) ? fx * fy                   : 0.0f;

            // clamped LDS indices (weight already 0 when clamped)
            const int cx0 = x0ok ? x0 : 0;
            const int cx1 = x1ok ? (x0 + 1) : 0;
            const int ry0 = y0ok ? (y0 << 7) : 0;
            const int ry1 = y1ok ? ((y0 + 1) << 7) : 0;

            rp[j] = img[ry0 + cx0] * w00 + img[ry0 + cx1] * w10 +
                    img[ry1 + cx0] * w01 + img[ry1 + cx1] * w11;
        }
        ovec[g] = r;   // 16-byte aligned, coalesced b128 store
    }
}

extern "C" void kernel_launch(void* const* d_in, const int* in_sizes, int n_in,
                              void* d_out, int out_size, void* d_ws, size_t ws_size,
                              hipStream_t stream) {
    const float* X     = (const float*)d_in[0];   // (16, 3, 128, 128) fp32
    const float* theta = (const float*)d_in[1];   // (16, 6) fp32
    float* out         = (float*)d_out;           // (16, 3, 128, 128) fp32

    const int nimg  = in_sizes[0] / NPIX;  // b * chan = 48
    const int batch = in_sizes[1] / 6;     // 16
    const int chan  = nimg / batch;        // 3

    dim3 grid(chan, batch);
    st_bilinear_kernel<<<grid, THREADS, 0, stream>>>(X, theta, out);
}